// NeuralTuringMachine_36979668418873
// MI455X (gfx1250) — compile-verified
//
#include <hip/hip_runtime.h>
#include <math.h>

#define BS      128
#define T_STEPS 32
#define DIN     128
#define DH      128
#define NS      512
#define EPS     1e-8f

typedef float v2f __attribute__((ext_vector_type(2)));
typedef float v8f __attribute__((ext_vector_type(8)));

__device__ __forceinline__ float sigmoidf_(float x) { return 1.f / (1.f + expf(-x)); }
__device__ __forceinline__ float softplusf_(float x) { return (x > 20.f) ? x : log1pf(expf(x)); }

// ---- WMMA f32 16x16x4 fragment loaders (layouts per CDNA5 ISA 7.12.2) ----
// A 16x4: lane&15 = row M; lane>>4 selects K-pair {0,1} vs {2,3}; VGPR0=K even, VGPR1=K odd
__device__ __forceinline__ v2f wmma_load_a(const float* A, int lda, int m0, int kk) {
  int l = threadIdx.x & 31;
  int row = m0 + (l & 15);
  int k = kk + ((l >> 4) << 1);
  v2f a; a.x = A[(size_t)row * lda + k]; a.y = A[(size_t)row * lda + k + 1]; return a;
}
// B 4x16: lane&15 = col N; half-lanes split K pairs, row-striped across lanes
__device__ __forceinline__ v2f wmma_load_b(const float* B, int ldb, int kk, int n0) {
  int l = threadIdx.x & 31;
  int col = n0 + (l & 15);
  int k = kk + ((l >> 4) << 1);
  v2f b; b.x = B[(size_t)k * ldb + col]; b.y = B[(size_t)(k + 1) * ldb + col]; return b;
}
__device__ __forceinline__ v2f wmma_load_b_guard(const float* B, int ldb, int kk, int n0, int N) {
  int l = threadIdx.x & 31;
  int col = n0 + (l & 15);
  int k = kk + ((l >> 4) << 1);
  v2f b;
  b.x = (col < N) ? B[(size_t)k * ldb + col] : 0.f;
  b.y = (col < N) ? B[(size_t)(k + 1) * ldb + col] : 0.f;
  return b;
}
#define WMMA_F32(a, b, c) \
  __builtin_amdgcn_wmma_f32_16x16x4_f32(false, (a), false, (b), (short)0, (c), false, false)

// =====================  K0: state init  =====================
__global__ void ntm_init(float* h0, float* r, float* w, float* mem, size_t memN) {
  size_t i = (size_t)blockIdx.x * blockDim.x + threadIdx.x;
  if (i < (size_t)BS * DH) { h0[i] = 0.f; r[i] = 0.f; }
  if (i < (size_t)BS * NS) w[i] = 1.f / (float)NS;
  if (i < memN) mem[i] = 1e-6f;
}

// =====================  K1: h = tanh(x_t@Wxh + h@Whh + r@Wrh + bh)  =====================
__global__ __launch_bounds__(32) void ntm_rnn_h(
    const float* __restrict__ x, const float* __restrict__ hA, const float* __restrict__ r,
    const float* __restrict__ Wxh, const float* __restrict__ Whh, const float* __restrict__ Wrh,
    const float* __restrict__ bh, float* __restrict__ hB, int t) {
  int n0 = blockIdx.x * 16, m0 = blockIdx.y * 16;
  const float* xt = x + (size_t)t * DIN;  // row b lives at b*T*DIN + t*DIN
  v8f c = {};
  for (int kk = 0; kk < DIN; kk += 4) {
    c = WMMA_F32(wmma_load_a(xt, T_STEPS * DIN, m0, kk), wmma_load_b(Wxh, DH, kk, n0), c);
    c = WMMA_F32(wmma_load_a(hA, DH, m0, kk),            wmma_load_b(Whh, DH, kk, n0), c);
    c = WMMA_F32(wmma_load_a(r,  DH, m0, kk),            wmma_load_b(Wrh, DH, kk, n0), c);
  }
  int l = threadIdx.x & 31;
  int col = n0 + (l & 15);
  int rb = m0 + ((l >> 4) << 3);
  float bias = bh[col];
#pragma unroll
  for (int i = 0; i < 8; ++i)
    hB[(size_t)(rb + i) * DH + col] = tanhf(c[i] + bias);
}

// =====================  K2: o_r = h@Wr + br ; o_w = h@Ww + bw  =====================
__global__ __launch_bounds__(32) void ntm_proj(
    const float* __restrict__ h,
    const float* __restrict__ Wr, const float* __restrict__ br,
    const float* __restrict__ Ww, const float* __restrict__ bw,
    float* __restrict__ o_r, float* __restrict__ o_w) {
  int nt = blockIdx.x, m0 = blockIdx.y * 16;
  const float* W; const float* bias; float* O; int N; int n0;
  if (nt < 9) { W = Wr; bias = br; O = o_r; N = DH + 6;     n0 = nt * 16; }
  else        { W = Ww; bias = bw; O = o_w; N = 3 * DH + 6; n0 = (nt - 9) * 16; }
  v8f c = {};
  for (int kk = 0; kk < DH; kk += 4)
    c = WMMA_F32(wmma_load_a(h, DH, m0, kk), wmma_load_b_guard(W, N, kk, n0, N), c);
  int l = threadIdx.x & 31;
  int col = n0 + (l & 15);
  int rb = m0 + ((l >> 4) << 3);
  if (col < N) {
    float bi = bias[col];
#pragma unroll
    for (int i = 0; i < 8; ++i)
      O[(size_t)(rb + i) * N + col] = c[i] + bi;
  }
}

// =====================  K4: y[:,t,:] = [h|r]@Wout + bout  =====================
__global__ __launch_bounds__(32) void ntm_out(
    const float* __restrict__ h, const float* __restrict__ r,
    const float* __restrict__ Wout, const float* __restrict__ bout,
    float* __restrict__ y, int t) {
  int n0 = blockIdx.x * 16, m0 = blockIdx.y * 16;
  v8f c = {};
  for (int kk = 0; kk < 2 * DH; kk += 4) {
    int l = threadIdx.x & 31;
    int row = m0 + (l & 15);
    int k = kk + ((l >> 4) << 1);
    v2f a;
    a.x = (k < DH)     ? h[(size_t)row * DH + k]     : r[(size_t)row * DH + k - DH];
    a.y = (k + 1 < DH) ? h[(size_t)row * DH + k + 1] : r[(size_t)row * DH + k + 1 - DH];
    c = WMMA_F32(a, wmma_load_b(Wout, DIN, kk, n0), c);
  }
  int l = threadIdx.x & 31;
  int col = n0 + (l & 15);
  int rb = m0 + ((l >> 4) << 3);
  float bi = bout[col];
#pragma unroll
  for (int i = 0; i < 8; ++i) {
    int b = rb + i;
    y[((size_t)b * T_STEPS + t) * DIN + col] = c[i] + bi;
  }
}

// =====================  block reductions (512 threads = 16 wave32)  =====================
__device__ __forceinline__ float blk_sum(float v, float* s_red) {
#pragma unroll
  for (int o = 16; o > 0; o >>= 1) v += __shfl_xor(v, o, 32);
  int wv = threadIdx.x >> 5;
  __syncthreads();
  if ((threadIdx.x & 31) == 0) s_red[wv] = v;
  __syncthreads();
  if (threadIdx.x == 0) { float t = 0.f; for (int i = 0; i < 16; ++i) t += s_red[i]; s_red[16] = t; }
  __syncthreads();
  return s_red[16];
}
__device__ __forceinline__ float blk_max(float v, float* s_red) {
#pragma unroll
  for (int o = 16; o > 0; o >>= 1) v = fmaxf(v, __shfl_xor(v, o, 32));
  int wv = threadIdx.x >> 5;
  __syncthreads();
  if ((threadIdx.x & 31) == 0) s_red[wv] = v;
  __syncthreads();
  if (threadIdx.x == 0) { float t = s_red[0]; for (int i = 1; i < 16; ++i) t = fmaxf(t, s_red[i]); s_red[16] = t; }
  __syncthreads();
  return s_red[16];
}

// =====================  NTM cosine attention head  =====================
__device__ void ntm_attention(const float* s_k, float beta, float g,
                              float s0, float s1, float s2, float gamma, float normk,
                              const float* memb, const float* wprev,
                              float* s_dot, float* s_nrm, float* s_wg,
                              float* s_wout, float* s_red) {
  int tid = threadIdx.x, l = tid & 31, wv = tid >> 5;
  float4 k4 = ((const float4*)s_k)[l];
  // dot(mem[n], k) and |mem[n]|^2, one wave per slot (coalesced 512B rows)
  for (int n = wv; n < NS; n += 16) {
    float4 m = ((const float4*)(memb + (size_t)n * DH))[l];
    float d = m.x * k4.x + m.y * k4.y + m.z * k4.z + m.w * k4.w;
    float q = m.x * m.x + m.y * m.y + m.z * m.z + m.w * m.w;
#pragma unroll
    for (int o = 16; o > 0; o >>= 1) { d += __shfl_xor(d, o, 32); q += __shfl_xor(q, o, 32); }
    if (l == 0) { s_dot[n] = d; s_nrm[n] = q; }
  }
  __syncthreads();
  // content softmax
  float sc = beta * s_dot[tid] / (sqrtf(s_nrm[tid]) * normk + EPS);
  float mx = blk_max(sc, s_red);
  float ex = expf(sc - mx);
  float sm = blk_sum(ex, s_red);
  float wc = ex / sm;
  // interpolate
  float wg = g * wc + (1.f - g) * wprev[tid];
  s_wg[tid] = wg;
  __syncthreads();
  // circular shift + sharpen + renormalize
  float wt = s0 * s_wg[(tid + 1) & (NS - 1)] + s1 * wg + s2 * s_wg[(tid + NS - 1) & (NS - 1)];
  float wp = powf(fmaxf(wt, 0.f), gamma);
  float ws = blk_sum(wp, s_red);
  s_wout[tid] = wp / (ws + EPS);
  __syncthreads();
}

// =====================  K3: per-batch attention + readout + memory update  =====================
__global__ __launch_bounds__(512) void ntm_step(
    const float* __restrict__ o_r, const float* __restrict__ o_w,
    float* __restrict__ mem, float* __restrict__ wprev, float* __restrict__ rout) {
  __shared__ float s_k[DH], s_e[DH], s_a[DH];
  __shared__ float s_dot[NS], s_nrm[NS], s_wg[NS], s_wr[NS], s_ww[NS];
  __shared__ float s_red[20];
  __shared__ float s_scal[8];
  __shared__ float s_rpart[16 * DH];

  int b = blockIdx.x;
  int tid = threadIdx.x, l = tid & 31, wv = tid >> 5;
  const float* orow  = o_r + (size_t)b * (DH + 6);
  const float* owrow = o_w + (size_t)b * (3 * DH + 6);
  float* memb = mem + (size_t)b * NS * DH;
  float* wpb  = wprev + (size_t)b * NS;

  // ---- read head params
  if (tid < DH) s_k[tid] = orow[tid];
  if (tid == 0) {
    s_scal[0] = softplusf_(orow[DH]);
    s_scal[1] = sigmoidf_(orow[DH + 1]);
    float a0 = orow[DH + 2], a1 = orow[DH + 3], a2 = orow[DH + 4];
    float m3 = fmaxf(a0, fmaxf(a1, a2));
    float e0 = expf(a0 - m3), e1 = expf(a1 - m3), e2 = expf(a2 - m3);
    float inv = 1.f / (e0 + e1 + e2);
    s_scal[2] = e0 * inv; s_scal[3] = e1 * inv; s_scal[4] = e2 * inv;
    s_scal[5] = 1.f + softplusf_(orow[DH + 5]);
  }
  __syncthreads();
  float nk = blk_sum((tid < DH) ? s_k[tid] * s_k[tid] : 0.f, s_red);
  float normk = sqrtf(nk);
  ntm_attention(s_k, s_scal[0], s_scal[1], s_scal[2], s_scal[3], s_scal[4], s_scal[5],
                normk, memb, wpb, s_dot, s_nrm, s_wg, s_wr, s_red);

  // ---- write head params (w_prev for this head is w_r)
  if (tid < DH) {
    s_k[tid] = owrow[tid];
    s_e[tid] = sigmoidf_(owrow[DH + 6 + tid]);
    s_a[tid] = tanhf(owrow[2 * DH + 6 + tid]);
  }
  if (tid == 0) {
    s_scal[0] = softplusf_(owrow[DH]);
    s_scal[1] = sigmoidf_(owrow[DH + 1]);
    float a0 = owrow[DH + 2], a1 = owrow[DH + 3], a2 = owrow[DH + 4];
    float m3 = fmaxf(a0, fmaxf(a1, a2));
    float e0 = expf(a0 - m3), e1 = expf(a1 - m3), e2 = expf(a2 - m3);
    float inv = 1.f / (e0 + e1 + e2);
    s_scal[2] = e0 * inv; s_scal[3] = e1 * inv; s_scal[4] = e2 * inv;
    s_scal[5] = 1.f + softplusf_(owrow[DH + 5]);
  }
  __syncthreads();
  nk = blk_sum((tid < DH) ? s_k[tid] * s_k[tid] : 0.f, s_red);
  normk = sqrtf(nk);
  ntm_attention(s_k, s_scal[0], s_scal[1], s_scal[2], s_scal[3], s_scal[4], s_scal[5],
                normk, memb, s_wr, s_dot, s_nrm, s_wg, s_ww, s_red);

  // ---- fused: r = w_r·mem(old) and mem = mem*(1 - w_w⊗e) + w_w⊗a (single pass)
  float4 e4 = ((const float4*)s_e)[l];
  float4 a4 = ((const float4*)s_a)[l];
  float4 racc = make_float4(0.f, 0.f, 0.f, 0.f);
  for (int n = wv; n < NS; n += 16) {
    float4* row = (float4*)(memb + (size_t)n * DH);
    float4 m = row[l];
    float wr = s_wr[n], ww = s_ww[n];
    racc.x += wr * m.x; racc.y += wr * m.y; racc.z += wr * m.z; racc.w += wr * m.w;
    m.x = m.x * (1.f - ww * e4.x) + ww * a4.x;
    m.y = m.y * (1.f - ww * e4.y) + ww * a4.y;
    m.z = m.z * (1.f - ww * e4.z) + ww * a4.z;
    m.w = m.w * (1.f - ww * e4.w) + ww * a4.w;
    row[l] = m;
  }
  float* rp = s_rpart + wv * DH + 4 * l;
  rp[0] = racc.x; rp[1] = racc.y; rp[2] = racc.z; rp[3] = racc.w;
  __syncthreads();
  if (tid < DH) {
    float acc = 0.f;
#pragma unroll
    for (int w = 0; w < 16; ++w) acc += s_rpart[w * DH + tid];
    rout[(size_t)b * DH + tid] = acc;
  }
  wpb[tid] = s_ww[tid];  // carry w <- w_w
}

// =====================  host-side launcher  =====================
extern "C" void kernel_launch(void* const* d_in, const int* in_sizes, int n_in,
                              void* d_out, int out_size, void* d_ws, size_t ws_size,
                              hipStream_t stream) {
  (void)in_sizes; (void)n_in; (void)out_size; (void)ws_size;
  const float* x    = (const float*)d_in[0];
  const float* Wxh  = (const float*)d_in[1];
  const float* Whh  = (const float*)d_in[2];
  const float* Wrh  = (const float*)d_in[3];
  const float* bh   = (const float*)d_in[4];
  const float* Wout = (const float*)d_in[5];
  const float* bout = (const float*)d_in[6];
  const float* Wr   = (const float*)d_in[7];
  const float* br   = (const float*)d_in[8];
  const float* Ww   = (const float*)d_in[9];
  const float* bw   = (const float*)d_in[10];
  float* y = (float*)d_out;

  float* ws = (float*)d_ws;
  float* h0  = ws;
  float* h1  = h0 + (size_t)BS * DH;
  float* rb  = h1 + (size_t)BS * DH;
  float* wp  = rb + (size_t)BS * DH;
  float* orr = wp + (size_t)BS * NS;
  float* oww = orr + (size_t)BS * (DH + 6);
  float* mem = oww + (size_t)BS * (3 * DH + 6);

  size_t memN = (size_t)BS * NS * DH;
  int initBlocks = (int)((memN + 255) / 256);
  ntm_init<<<initBlocks, 256, 0, stream>>>(h0, rb, wp, mem, memN);

  for (int t = 0; t < T_STEPS; ++t) {
    const float* hA = (t & 1) ? h1 : h0;
    float* hB = (t & 1) ? h0 : h1;
    ntm_rnn_h<<<dim3(DH / 16, BS / 16), 32, 0, stream>>>(x, hA, rb, Wxh, Whh, Wrh, bh, hB, t);
    ntm_proj<<<dim3(34, BS / 16), 32, 0, stream>>>(hB, Wr, br, Ww, bw, orr, oww);
    ntm_step<<<BS, 512, 0, stream>>>(orr, oww, mem, wp, rb);
    ntm_out<<<dim3(DIN / 16, BS / 16), 32, 0, stream>>>(hB, rb, Wout, bout, y, t);
  }
}